// CirculantFeedForward_1589137900083
// MI455X (gfx1250) — compile-verified
//
#include <hip/hip_runtime.h>
#include <hip/hip_bf16.h>

// ---------------------------------------------------------------------------
// Block-circulant SwiGLU FFN (CirCNN) for MI455X / gfx1250.
// Frequency-domain algorithm: per 8-token tile, everything lives in LDS
// (320 KB / WGP). Down-projection einsum runs on V_WMMA_F32_16X16X4_F32.
// x tiles are moved global->LDS with GLOBAL_LOAD_ASYNC_TO_LDS_B128 (ASYNCcnt).
// ---------------------------------------------------------------------------

typedef float v2f __attribute__((ext_vector_type(2)));
typedef float v8f __attribute__((ext_vector_type(8)));

#define TOK 8           // tokens per workgroup tile
#define BLK 512         // circulant block size
#define NF  257         // rfft bins
#define DMODEL 2048
#define PIN 4           // d_model / BLK
#define QFF 11          // d_ff / BLK

// LDS layout (float2 units)
#define OFF_TW     0
#define OFF_XF     (OFF_TW  + 256)          // [4][TOK][257]
#define OFF_YF     (OFF_XF  + 4*TOK*NF)     // [4][TOK][257]
#define OFF_HFC    (OFF_YF  + 4*TOK*NF)     // [4 qslots][TOK][257]
#define OFF_WS0    (OFF_HFC + 4*TOK*NF)     // [TOK][512]
#define OFF_WS1    (OFF_WS0 + TOK*BLK)
#define OFF_GQ     (OFF_WS1 + TOK*BLK)      // [TOK][257]
#define OFF_UQ     (OFF_GQ  + TOK*NF)
#define OFF_XSTAGE (OFF_UQ  + TOK*NF)       // [TOK][512] floats (async staging)
#define SMEM_F2    (OFF_XSTAGE + (TOK*BLK)/2)   // total float2 count
// SMEM_F2 = 37232 + 2048 = 39280 float2 -> 314,240 bytes < 320 KB

#if defined(__gfx1250__)
#define ASYNC_LDS 1
#endif

__device__ __forceinline__ float2 cmul(float2 a, float2 b) {
    return make_float2(fmaf(a.x, b.x, -a.y * b.y), fmaf(a.x, b.y, a.y * b.x));
}
__device__ __forceinline__ float2 cadd(float2 a, float2 b) { return make_float2(a.x + b.x, a.y + b.y); }
__device__ __forceinline__ float2 csub(float2 a, float2 b) { return make_float2(a.x - b.x, a.y - b.y); }
__device__ __forceinline__ int brev9(int j) { return (int)(__brev((unsigned)j) >> 23); }

// In-place radix-2 DIF, natural input -> bit-reversed output, twiddles e^{-i}
// Executed cooperatively by ONE wave32 on its private 512-entry LDS region:
// LDS ops from the same wave complete in order -> no barriers needed.
__device__ __forceinline__ void fft512_dif(float2* buf, const float2* tw, int lane) {
    #pragma unroll
    for (int s = 0; s < 9; ++s) {
        const int half  = 256 >> s;        // m = half*2
        const int tstep = 1 << s;          // 512/m
        for (int idx = lane; idx < 256; idx += 32) {
            const int blk = idx >> (8 - s);
            const int j   = idx & (half - 1);
            const int a   = blk * (half << 1) + j;
            const int b   = a + half;
            float2 u = buf[a], v = buf[b];
            buf[a] = cadd(u, v);
            buf[b] = cmul(csub(u, v), tw[j * tstep]);
        }
    }
}

// In-place radix-2 DIT inverse, bit-reversed input -> natural output (unscaled)
__device__ __forceinline__ void fft512_dit_inv(float2* buf, const float2* tw, int lane) {
    #pragma unroll
    for (int s = 0; s < 9; ++s) {
        const int half  = 1 << s;
        const int tstep = 256 >> s;
        for (int idx = lane; idx < 256; idx += 32) {
            const int blk = idx >> s;
            const int j   = idx & (half - 1);
            const int a   = blk * (half << 1) + j;
            const int b   = a + half;
            float2 w = tw[j * tstep];
            w.y = -w.y;                    // e^{+i}
            float2 t = cmul(buf[b], w);
            float2 u = buf[a];
            buf[a] = cadd(u, t);
            buf[b] = csub(u, t);
        }
    }
}

// Fill FFT workspace with hermitian-extended spectrum in bit-reversed order
__device__ __forceinline__ void fill_herm_bitrev(float2* buf, const float2* spec, int lane) {
    for (int j = lane; j < 512; j += 32) {
        const int k = brev9(j);
        float2 sv;
        if (k <= 256) sv = spec[k];
        else { sv = spec[512 - k]; sv.y = -sv.y; }
        buf[j] = sv;
    }
}

// ---------------------------------------------------------------------------
// Init kernels: weight spectra into workspace (naive DFT; 132 rows, tiny)
// Layouts:
//   Wgf/Wuf: [q][f][p] float2   (VALU gate/up einsum reads p contiguously)
//   Wdf:     [f][chunk3][k4][m16] float2, zero padded (WMMA A-operand layout)
// ---------------------------------------------------------------------------
__global__ void wf_zero(float* p, int n) {
    for (int i = blockIdx.x * blockDim.x + threadIdx.x; i < n; i += gridDim.x * blockDim.x)
        p[i] = 0.0f;
}

__global__ void wf_init(const float* __restrict__ wg, const float* __restrict__ wu,
                        const float* __restrict__ wd,
                        float2* __restrict__ Wgf, float2* __restrict__ Wuf,
                        float2* __restrict__ Wdf) {
    const int v = blockIdx.x;              // 0..131 circulant vectors
    const float* src;
    int cse, q, p;
    if (v < 44)      { src = wg + v * BLK;        q = v / 4;        p = v % 4;        cse = 0; }
    else if (v < 88) { int r = v - 44; src = wu + r * BLK; q = r / 4; p = r % 4;      cse = 1; }
    else             { int r = v - 88; src = wd + r * BLK; p = r / 11; q = r % 11;    cse = 2; }

    __shared__ float s[BLK];
    for (int i = threadIdx.x; i < BLK; i += blockDim.x) s[i] = src[i];
    __syncthreads();

    const float c0 = -6.283185307179586f / 512.0f;
    for (int f = threadIdx.x; f <= 256; f += blockDim.x) {
        float re = 0.0f, im = 0.0f;
        for (int n = 0; n < BLK; ++n) {
            float ang = c0 * (float)((f * n) & 511);
            float sn, cn;
            sincosf(ang, &sn, &cn);
            re = fmaf(s[n], cn, re);
            im = fmaf(s[n], sn, im);
        }
        float2 val = make_float2(re, im);
        if (cse == 0)      Wgf[(q * NF + f) * 4 + p] = val;
        else if (cse == 1) Wuf[(q * NF + f) * 4 + p] = val;
        else {
            const int c = q >> 2, k = q & 3;
            Wdf[((f * 3 + c) * 4 + k) * 16 + p] = val;   // rest stays zero
        }
    }
}

// ---------------------------------------------------------------------------
// Main fused kernel: one 8-token tile per workgroup, 256 threads = 8 waves.
// ---------------------------------------------------------------------------
__global__ __launch_bounds__(256) void circffn_main(
    const float* __restrict__ x, const float2* __restrict__ Wgf,
    const float2* __restrict__ Wuf, const float2* __restrict__ Wdf,
    float* __restrict__ out)
{
    extern __shared__ float2 smem[];
    float2* tw     = smem + OFF_TW;
    float2* Xf     = smem + OFF_XF;
    float2* Yf     = smem + OFF_YF;
    float2* Hfc    = smem + OFF_HFC;
    float2* ws0    = smem + OFF_WS0;
    float2* ws1    = smem + OFF_WS1;
    float2* Gq     = smem + OFF_GQ;
    float2* Uq     = smem + OFF_UQ;
    float*  xstage = (float*)(smem + OFF_XSTAGE);

    const int tid  = threadIdx.x;
    const int lane = tid & 31;
    const int wave = tid >> 5;             // == token within tile for FFT phases
    const int tok0 = blockIdx.x * TOK;
    const float inv512 = 1.0f / 512.0f;

    // twiddles: tw[t] = e^{-2pi i t/512}; zero the Yf accumulator
    {
        float th = 6.283185307179586f * (float)tid * (1.0f / 512.0f);
        float sn, cn; sincosf(th, &sn, &cn);
        tw[tid] = make_float2(cn, -sn);
    }
    for (int i = tid; i < 4 * TOK * NF; i += 256) Yf[i] = make_float2(0.f, 0.f);
    __syncthreads();

    // ---- Phase 1: forward FFT of the 4 input blocks -> Xf[p][tok][f] ----
    for (int p = 0; p < PIN; ++p) {
        // global -> LDS staging, 16B per lane per issue (ASYNCcnt-tracked DMA)
        for (int i = tid; i < (TOK * BLK) / 4; i += 256) {
            const int t  = i >> 7;               // 128 float4 per token row
            const int n4 = (i & 127) << 2;
            const float* gp = &x[(size_t)(tok0 + t) * DMODEL + p * BLK + n4];
#ifdef ASYNC_LDS
            const unsigned loff = __builtin_amdgcn_groupstaticsize()
                                + (unsigned)(OFF_XSTAGE * sizeof(float2))
                                + (unsigned)((t * BLK + n4) * sizeof(float));
            asm volatile("global_load_async_to_lds_b128 %0, %1, off"
                         :: "v"(loff), "v"((unsigned long long)(size_t)gp)
                         : "memory");
#else
            ((float4*)&xstage[t * BLK + n4])[0] = ((const float4*)gp)[0];
#endif
            if (p < PIN - 1) __builtin_prefetch(gp + BLK, 0, 3);
        }
#ifdef ASYNC_LDS
        asm volatile("s_wait_asynccnt 0" ::: "memory");  // this wave's DMAs done
#endif
        __syncthreads();                                 // all waves' DMAs done
        // expand to complex FFT workspace
        for (int i = tid; i < TOK * BLK; i += 256)
            ws0[i] = make_float2(xstage[i], 0.0f);
        __syncthreads();

        float2* buf = ws0 + wave * BLK;
        fft512_dif(buf, tw, lane);
        for (int k = lane; k <= 256; k += 32)
            Xf[(p * TOK + wave) * NF + k] = buf[brev9(k)];
        __syncthreads();
    }

    // ---- Phase 2: q loop (gate/up einsum, irfft, SwiGLU, rfft(h), WMMA down) ----
    for (int q = 0; q < QFF; ++q) {
        // gate/up frequency einsum over p (VALU; only ~3 GFLOP total)
        for (int i = tid; i < TOK * NF; i += 256) {
            const int t = i & 7, f = i >> 3;
            float2 ag = make_float2(0.f, 0.f), au = make_float2(0.f, 0.f);
            #pragma unroll
            for (int p = 0; p < PIN; ++p) {
                const float2 xv = Xf[(p * TOK + t) * NF + f];
                ag = cadd(ag, cmul(xv, Wgf[(q * NF + f) * 4 + p]));
                au = cadd(au, cmul(xv, Wuf[(q * NF + f) * 4 + p]));
            }
            Gq[t * NF + f] = ag;
            Uq[t * NF + f] = au;
        }
        __syncthreads();

        // wave-private: irfft(gate), irfft(up), h = silu(g)*u, rfft(h)
        {
            float2* bg = ws0 + wave * BLK;
            float2* bu = ws1 + wave * BLK;
            fill_herm_bitrev(bg, Gq + wave * NF, lane);
            fill_herm_bitrev(bu, Uq + wave * NF, lane);
            fft512_dit_inv(bg, tw, lane);
            fft512_dit_inv(bu, tw, lane);
            for (int n = lane; n < BLK; n += 32) {
                const float g = bg[n].x * inv512;
                const float u = bu[n].x * inv512;
                const float h = (g / (1.0f + __expf(-g))) * u;   // SiLU * up
                bg[n] = make_float2(h, 0.0f);
            }
            fft512_dif(bg, tw, lane);
            const int slot = q & 3;
            for (int k = lane; k <= 256; k += 32)
                Hfc[(slot * TOK + wave) * NF + k] = bg[brev9(k)];
        }

        // chunk of 4 q's complete -> accumulate Yf += Wd_chunk x Hf_chunk
        if (q == 3 || q == 7 || q == 10) {
            __syncthreads();
            const int c  = q >> 2;
            const int m  = lane & 15;           // A row (p), rows >=4 are zero
            const int s0 = (lane >> 4) * 2;     // K-half base (q within chunk)
            const int n  = lane & 15;           // B/D column (token)
            const int t  = n & 7;
            const bool nvalid = (n < 8);
            for (int f = wave; f < NF; f += 8) {   // waves partition frequencies
                const float2 A0 = Wdf[((f * 3 + c) * 4 + s0    ) * 16 + m];
                const float2 A1 = Wdf[((f * 3 + c) * 4 + s0 + 1) * 16 + m];
                float2 B0 = Hfc[((s0    ) * TOK + t) * NF + f];
                float2 B1 = Hfc[((s0 + 1) * TOK + t) * NF + f];
                B0.x = nvalid ? B0.x : 0.f;  B0.y = nvalid ? B0.y : 0.f;
                B1.x = nvalid ? B1.x : 0.f;  B1.y = nvalid ? B1.y : 0.f;
#if __has_builtin(__builtin_amdgcn_wmma_f32_16x16x4_f32)
                v2f awr  = {A0.x, A1.x};
                v2f awi  = {A0.y, A1.y};
                v2f awin = {-A0.y, -A1.y};
                v2f bhr  = {B0.x, B1.x};
                v2f bhi  = {B0.y, B1.y};
                v8f cr = {0,0,0,0,0,0,0,0}, ci = {0,0,0,0,0,0,0,0};
                if (lane < 8) {
                    #pragma unroll
                    for (int r = 0; r < 4; ++r) {
                        const float2 yv = Yf[(r * TOK + lane) * NF + f];
                        cr[r] = yv.x; ci[r] = yv.y;
                    }
                }
                // Yr += Wr*Hr - Wi*Hi ; Yi += Wi*Hr + Wr*Hi  (4 chained WMMAs)
                cr = __builtin_amdgcn_wmma_f32_16x16x4_f32(false, awr,  false, bhr, (short)0, cr, false, false);
                cr = __builtin_amdgcn_wmma_f32_16x16x4_f32(false, awin, false, bhi, (short)0, cr, false, false);
                ci = __builtin_amdgcn_wmma_f32_16x16x4_f32(false, awi,  false, bhr, (short)0, ci, false, false);
                ci = __builtin_amdgcn_wmma_f32_16x16x4_f32(false, awr,  false, bhi, (short)0, ci, false, false);
                if (lane < 8) {
                    #pragma unroll
                    for (int r = 0; r < 4; ++r)
                        Yf[(r * TOK + lane) * NF + f] = make_float2(cr[r], ci[r]);
                }
#else
                // VALU fallback: only lanes 0..3 pass, each owns one (p=m, tok=t=m) pair
                if (m < 4 && nvalid && lane < 16) {
                    float2 acc = Yf[(m * TOK + t) * NF + f];
                    acc = cadd(acc, cmul(A0, B0));
                    acc = cadd(acc, cmul(A1, B1));
                    const float2 A2 = Wdf[((f * 3 + c) * 4 + 2) * 16 + m];
                    const float2 A3 = Wdf[((f * 3 + c) * 4 + 3) * 16 + m];
                    const float2 B2 = Hfc[(2 * TOK + t) * NF + f];
                    const float2 B3 = Hfc[(3 * TOK + t) * NF + f];
                    if (s0 == 0) {
                        acc = cadd(acc, cmul(A2, B2));
                        acc = cadd(acc, cmul(A3, B3));
                        Yf[(m * TOK + t) * NF + f] = acc;
                    }
                }
#endif
            }
        }
        __syncthreads();
    }

    // ---- Phase 3: inverse FFT of Yf -> output (wave-private per token) ----
    {
        float2* buf = ws0 + wave * BLK;
        for (int p = 0; p < PIN; ++p) {
            fill_herm_bitrev(buf, Yf + (p * TOK + wave) * NF, lane);
            fft512_dit_inv(buf, tw, lane);
            for (int nn = lane; nn < BLK; nn += 32)
                out[(size_t)(tok0 + wave) * DMODEL + p * BLK + nn] = buf[nn].x * inv512;
        }
    }
}

// ---------------------------------------------------------------------------
// Workspace layout (floats):
//   Wgf: [11][257][4] cplx  = 22616 floats
//   Wuf: same               = 22616 floats
//   Wdf: [257][3][4][16] cplx = 98688 floats   (zero padded for WMMA)
// ---------------------------------------------------------------------------
#define WGF_FLOATS (QFF * NF * 4 * 2)
#define WDF_FLOATS (NF * 3 * 4 * 16 * 2)

extern "C" void kernel_launch(void* const* d_in, const int* in_sizes, int n_in,
                              void* d_out, int out_size, void* d_ws, size_t ws_size,
                              hipStream_t stream) {
    const float* x  = (const float*)d_in[0];
    const float* wg = (const float*)d_in[1];
    const float* wu = (const float*)d_in[2];
    const float* wd = (const float*)d_in[3];
    float* out = (float*)d_out;
    float* ws  = (float*)d_ws;

    float2* Wgf = (float2*)ws;
    float2* Wuf = (float2*)(ws + WGF_FLOATS);
    float2* Wdf = (float2*)(ws + 2 * WGF_FLOATS);

    // weight spectra (tiny, recomputed every launch; deterministic)
    wf_zero<<<208, 256, 0, stream>>>((float*)Wdf, WDF_FLOATS);
    wf_init<<<132, 256, 0, stream>>>(wg, wu, wd, Wgf, Wuf, Wdf);

    const int smem_bytes = SMEM_F2 * (int)sizeof(float2);   // 314,240 B < 320 KB
    (void)hipFuncSetAttribute(reinterpret_cast<const void*>(circffn_main),
                              hipFuncAttributeMaxDynamicSharedMemorySize, smem_bytes);

    const int ntok = in_sizes[0] / DMODEL;                   // 16384
    circffn_main<<<ntok / TOK, 256, smem_bytes, stream>>>(x, Wgf, Wuf, Wdf, out);
}